// ChampherLoss_37623913513196
// MI455X (gfx1250) — compile-verified
//
#include <hip/hip_runtime.h>

// ---------------------------------------------------------------------------
// Chamfer loss on MI455X (gfx1250, wave32).
//
// dist(n,m) = |x_n|^2 + |y_m|^2 - 2 x_n.y_m
//
// V_WMMA_F32_16X16X4_F32 with the K=4 pad slot used to fold the affine term:
//   A row m  = (-2x0, -2x1 | -2x2, 1.0)
//   B col n  = ( y0,   y1  |  y2,  |y|^2)
//   => D[m][n] = -2 x.y + |y|^2      (|x|^2 added once per row at the end)
//
// Each wave owns a 16-row strip and sweeps all 256 column tiles keeping
// per-lane running mins -> no atomics; second pass swaps the clouds.
// All workspace data (~2.5 MB) is L2-resident; the loop is issue-bound: per 16x16
// tile it is 1 b64 load + 1 wmma + 8 v_min (min3-paired by the compiler).
// ---------------------------------------------------------------------------

typedef __attribute__((ext_vector_type(2))) float v2f;
typedef __attribute__((ext_vector_type(8))) float v8f;

#define BATCH 8
#define NPTS  4096          // N == M == 4096
#define TILES (NPTS / 16)   // 256 tiles of 16 points per batch

// ---------------------------------------------------------------------------
// Pack a point cloud (B, NPTS, 3) f32 into WMMA-ready operand tiles, in both
// the A-role and B-role forms, plus squared norms.
// Per tile t, lane l (wave32): point p = t*16 + (l & 15), half = l >> 4.
//   A-form: half 0 -> (-2x, -2y)   half 1 -> (-2z, 1.0)
//   B-form: half 0 -> ( x,   y )   half 1 -> ( z, |p|^2)
// (A layout: lanes 0-15 hold K=0,1; lanes 16-31 hold K=2,3. B mirrored.)
// ---------------------------------------------------------------------------
__global__ void chamfer_pack_kernel(const float* __restrict__ pts,
                                    float* __restrict__ packedA,
                                    float* __restrict__ packedB,
                                    float* __restrict__ norms) {
    int gid   = blockIdx.x * blockDim.x + threadIdx.x;   // over BATCH*TILES*32
    int lane  = gid & 31;
    int tg    = gid >> 5;            // b*TILES + tile
    int b     = tg >> 8;             // TILES == 256
    int tile  = tg & (TILES - 1);
    int p     = tile * 16 + (lane & 15);
    int half  = lane >> 4;

    const float* q = pts + ((size_t)b * NPTS + p) * 3;
    float q0 = q[0], q1 = q[1], q2 = q[2];
    float n2 = q0 * q0 + q1 * q1 + q2 * q2;

    v2f ra, rb;
    if (half) {
        ra.x = -2.0f * q2;  ra.y = 1.0f;
        rb.x = q2;          rb.y = n2;
    } else {
        ra.x = -2.0f * q0;  ra.y = -2.0f * q1;
        rb.x = q0;          rb.y = q1;
    }
    size_t slot = (size_t)tg * 32 + lane;
    ((v2f*)packedA)[slot] = ra;
    ((v2f*)packedB)[slot] = rb;

    if (half == 0)
        norms[(size_t)b * NPTS + p] = n2;
}

// ---------------------------------------------------------------------------
// One wave per (b, rowTile); sweep all column tiles.
// D layout: D[v], lane l -> row = v + 8*(l>=16), col = l & 15.
// EXEC stays all-ones through the WMMA loop (uniform control flow).
// ---------------------------------------------------------------------------
__global__ void chamfer_min_kernel(const float* __restrict__ packedA,
                                   const float* __restrict__ packedB,
                                   const float* __restrict__ rowNorms,
                                   float* __restrict__ rowMin) {
    int lane    = threadIdx.x & 31;
    int wave    = (blockIdx.x * blockDim.x + threadIdx.x) >> 5;  // strip id
    int b       = wave >> 8;                 // TILES == 256
    int rowTile = wave & (TILES - 1);

    const v2f* pA = (const v2f*)packedA;
    const v2f* pB = (const v2f*)packedB + (size_t)b * TILES * 32;

    // A operand: fixed for the whole sweep.
    v2f a = pA[((size_t)b * TILES + rowTile) * 32 + lane];

    float rmin[8];
#pragma unroll
    for (int v = 0; v < 8; ++v) rmin[v] = 3.0e38f;

    for (int ct = 0; ct < TILES; ++ct) {
        v2f bb = pB[ct * 32 + lane];

        v8f c = {};
        // D = (-2x) . y + |y|^2 for a full 16x16 tile, in one instruction.
        c = __builtin_amdgcn_wmma_f32_16x16x4_f32(
                /*neg_a=*/false, a, /*neg_b=*/false, bb,
                /*c_mod=*/(short)0, c, /*reuse_a=*/false, /*reuse_b=*/false);

#pragma unroll
        for (int v = 0; v < 8; ++v)
            rmin[v] = fminf(rmin[v], c[v]);
    }

    // Min across the 16 lanes of each half (cols live in lanes; rows 0..7 in
    // half 0, rows 8..15 in half 1). xor masks 1,2,4,8 stay within 16-groups.
#pragma unroll
    for (int off = 1; off < 16; off <<= 1) {
#pragma unroll
        for (int v = 0; v < 8; ++v)
            rmin[v] = fminf(rmin[v], __shfl_xor(rmin[v], off, 32));
    }

    if ((lane & 15) == 0) {                  // lanes 0 and 16 write 8 rows each
        int rowbase = rowTile * 16 + ((lane >> 4) ? 8 : 0);
#pragma unroll
        for (int v = 0; v < 8; ++v) {
            int r = rowbase + v;
            rowMin[(size_t)b * NPTS + r] =
                rmin[v] + rowNorms[(size_t)b * NPTS + r];
        }
    }
}

// ---------------------------------------------------------------------------
// mean(dist1) + mean(dist2), single block of 256 threads over 2x32K floats.
// ---------------------------------------------------------------------------
__global__ void chamfer_reduce_kernel(const float* __restrict__ dist1,
                                      const float* __restrict__ dist2,
                                      float* __restrict__ out) {
    __shared__ float sm[256];
    const int n   = BATCH * NPTS;
    const float w = 1.0f / (float)n;
    int tid = threadIdx.x;

    float acc = 0.0f;
    for (int i = tid; i < n; i += 256) acc += dist1[i] * w;
    for (int i = tid; i < n; i += 256) acc += dist2[i] * w;

    sm[tid] = acc;
    __syncthreads();
    for (int s = 128; s > 0; s >>= 1) {
        if (tid < s) sm[tid] += sm[tid + s];
        __syncthreads();
    }
    if (tid == 0) out[0] = sm[0];
}

extern "C" void kernel_launch(void* const* d_in, const int* in_sizes, int n_in,
                              void* d_out, int out_size, void* d_ws, size_t ws_size,
                              hipStream_t stream) {
    (void)in_sizes; (void)n_in; (void)out_size; (void)ws_size;

    const float* x = (const float*)d_in[0];   // receptive_pc (B, N, 3) f32
    const float* y = (const float*)d_in[1];   // decoder_pc   (B, M, 3) f32
    float* out = (float*)d_out;               // scalar f32

    // Workspace carve-up (floats): 4 packed operand arrays + norms + mins.
    const size_t PACKSZ = (size_t)BATCH * TILES * 64;  // 131072 floats each
    const size_t VECSZ  = (size_t)BATCH * NPTS;        // 32768 floats each
    float* ws       = (float*)d_ws;
    float* packedXA = ws;
    float* packedXB = packedXA + PACKSZ;
    float* packedYA = packedXB + PACKSZ;
    float* packedYB = packedYA + PACKSZ;
    float* x2       = packedYB + PACKSZ;
    float* y2       = x2 + VECSZ;
    float* dist1    = y2 + VECSZ;
    float* dist2    = dist1 + VECSZ;

    dim3 blk(256);
    dim3 gridPack(BATCH * TILES * 32 / 256);            // 256 blocks
    chamfer_pack_kernel<<<gridPack, blk, 0, stream>>>(x, packedXA, packedXB, x2);
    chamfer_pack_kernel<<<gridPack, blk, 0, stream>>>(y, packedYA, packedYB, y2);

    // 2048 waves per pass, 8 waves (256 threads) per block -> 256 blocks.
    dim3 gridMin(BATCH * TILES * 32 / 256);
    // Pass 1: rows = x points, mins over y  -> dist1 (B, N)
    chamfer_min_kernel<<<gridMin, blk, 0, stream>>>(packedXA, packedYB, x2, dist1);
    // Pass 2: rows = y points, mins over x  -> dist2 (B, M)
    chamfer_min_kernel<<<gridMin, blk, 0, stream>>>(packedYA, packedXB, y2, dist2);

    chamfer_reduce_kernel<<<1, blk, 0, stream>>>(dist1, dist2, out);
}